// PT_DWT_38053410242636
// MI455X (gfx1250) — compile-verified
//
#include <hip/hip_runtime.h>
#include <cstdint>
#include <cstddef>

// Haar DWT (stride-1 depthwise 2x2, 4 subbands) for x[8,64,256,256] f32.
// Memory-bound: ~128MiB in + ~512MiB out -> ~29us floor @ 23.3 TB/s.
// Strategy: TDM tensor_load_to_lds stages a 64x256 f32 row-band (64KB LDS)
// per workgroup; compute is a trivial 4-tap butterfly; output uses
// non-temporal B128 stores (pure streaming, 4x the read volume).

static constexpr int W  = 256;   // image width (floats)
static constexpr int H  = 256;   // image height
static constexpr int TB = 64;    // output rows per workgroup band
static constexpr int BANDS = H / TB;  // 4

typedef float    f32x4 __attribute__((ext_vector_type(4)));
typedef uint32_t u32x4 __attribute__((ext_vector_type(4)));
typedef int32_t  i32x4 __attribute__((ext_vector_type(4)));
typedef int32_t  i32x8 __attribute__((ext_vector_type(8)));

__global__ __launch_bounds__(256) void PT_DWT_haar_kernel(const float* __restrict__ x,
                                                          float* __restrict__ out) {
    __shared__ __align__(16) float tile[TB * W];   // 64 KB: rows r0..r0+63

    const int bid = blockIdx.x;
    const int img = bid >> 2;                  // b*C + c  (0..511)
    const int r0  = (bid & (BANDS - 1)) * TB;  // first output row of this band
    const float* __restrict__ src = x + (size_t)img * (H * W) + (size_t)r0 * W;

#if defined(__gfx1250__) && __has_builtin(__builtin_amdgcn_tensor_load_to_lds) && __has_builtin(__builtin_amdgcn_s_wait_tensorcnt)
    // --- TDM path: one 2D tensor DMA (64 rows x 256 cols x 4B) into LDS ---
    if (threadIdx.x < 32) {   // wave 0 only: TDM issues once per wave, EXEC-independent
        const uint64_t ga = (uint64_t)(uintptr_t)src;
        const uint32_t lds_off = (uint32_t)(uintptr_t)(void*)&tile[0]; // low 32b = LDS offset

        u32x4 g0;
        g0[0] = 1u;                                   // count=1, user descriptor, no gather
        g0[1] = lds_off;                              // lds_addr [63:32]
        g0[2] = (uint32_t)ga;                         // global_addr [95:64]
        g0[3] = ((uint32_t)(ga >> 32) & 0x01FFFFFFu)  // global_addr [120:96]
                | (2u << 30);                         // type=2 ("image")

        i32x8 g1;
        g1[0] = (int32_t)(2u << 16);                  // workgroup_mask=0, data_size=2 (4B)
        g1[1] = (int32_t)((uint32_t)W << 16);         // tensor_dim0 lo16 (=256) at bits[63:48]
        g1[2] = (int32_t)((uint32_t)TB << 16);        // tensor_dim0 hi16=0 | tensor_dim1 lo16 (=64)
        g1[3] = (int32_t)((uint32_t)W << 16);         // tensor_dim1 hi16=0 | tile_dim0 (=256)
        g1[4] = (int32_t)TB;                          // tile_dim1=64, tile_dim2=0 (2D)
        g1[5] = (int32_t)W;                           // tensor_dim0_stride lo32 (row pitch, elems)
        g1[6] = 0;                                    // stride0 hi | stride1 lo (unused for 2D)
        g1[7] = 0;

        i32x4 gz4 = {0, 0, 0, 0};                     // groups 2/3 unused (2D tensor)
        i32x8 gz8 = {0, 0, 0, 0, 0, 0, 0, 0};         // trailing group (clang-23 6-arg form)
        __builtin_amdgcn_tensor_load_to_lds(g0, g1, gz4, gz4, gz8, 0);
        __builtin_amdgcn_s_wait_tensorcnt(0);
    }
    __syncthreads();
#else
    // --- fallback: cooperative B128 fills ---
    for (int i = threadIdx.x; i < TB * (W / 4); i += 256) {
        ((f32x4*)tile)[i] = ((const f32x4*)src)[i];
    }
    __syncthreads();
#endif

    const int tid  = threadIdx.x;
    const int w    = (tid & 63) * 4;        // 64 lanes cover a 256-float row
    const bool edge = (w + 4 >= W);         // right zero-pad (w==252)
    float* const oimg  = out + (size_t)img * 4 * (H * W);
    const size_t plane = (size_t)H * W;

    for (int lr = (tid >> 6); lr < TB; lr += 4) {
        // row h = r0+lr and row h+1 (with bottom zero-pad at h==H-1)
        const f32x4 a  = *(const f32x4*)&tile[lr * W + w];
        const float a4 = edge ? 0.0f : tile[lr * W + w + 4];

        f32x4 b  = {0.0f, 0.0f, 0.0f, 0.0f};
        float b4 = 0.0f;
        if (lr < TB - 1) {
            b  = *(const f32x4*)&tile[(lr + 1) * W + w];
            b4 = edge ? 0.0f : tile[(lr + 1) * W + w + 4];
        } else if (r0 + TB < H) {
            // halo row from global (coalesced, L2-cached; 1/64 of the reads)
            const float* g = src + (size_t)TB * W + w;
            b  = *(const f32x4*)g;
            b4 = edge ? 0.0f : g[4];
        } // else: h == H-1 -> padded zeros

        const float t00[4] = {a.x, a.y, a.z, a.w};
        const float t01[4] = {a.y, a.z, a.w, a4};
        const float t10[4] = {b.x, b.y, b.z, b.w};
        const float t11[4] = {b.y, b.z, b.w, b4};

        f32x4 ll, lh, hl, hh;
#pragma unroll
        for (int j = 0; j < 4; ++j) {
            const float s0 = t00[j] + t01[j];   // row h sum
            const float d0 = t00[j] - t01[j];   // row h diff
            const float s1 = t10[j] + t11[j];   // row h+1 sum
            const float d1 = t10[j] - t11[j];   // row h+1 diff
            ll[j] = 0.5f * (s0 + s1);
            lh[j] = 0.5f * (s0 - s1);
            hl[j] = 0.5f * (d0 + d1);
            hh[j] = 0.5f * (d0 - d1);
        }

        const size_t o = (size_t)(r0 + lr) * W + w;
        __builtin_nontemporal_store(ll, (f32x4*)(oimg + o));                // LL -> 4c+0
        __builtin_nontemporal_store(lh, (f32x4*)(oimg + plane + o));        // LH -> 4c+1
        __builtin_nontemporal_store(hl, (f32x4*)(oimg + 2 * plane + o));    // HL -> 4c+2
        __builtin_nontemporal_store(hh, (f32x4*)(oimg + 3 * plane + o));    // HH -> 4c+3
    }
}

extern "C" void kernel_launch(void* const* d_in, const int* in_sizes, int n_in,
                              void* d_out, int out_size, void* d_ws, size_t ws_size,
                              hipStream_t stream) {
    const float* x = (const float*)d_in[0];
    float* out = (float*)d_out;
    const int nimg = in_sizes[0] / (H * W);   // B*C = 512
    dim3 grid(nimg * BANDS);                  // 2048 workgroups
    dim3 block(256);                          // 8 wave32 waves
    PT_DWT_haar_kernel<<<grid, block, 0, stream>>>(x, out);
}